// GATDropModel_11132555231939
// MI455X (gfx1250) — compile-verified
//
#include <hip/hip_runtime.h>
#include <math.h>

// ---- problem constants -------------------------------------------------
#define NN     100                 // nodes
#define HH     12                  // heads
#define NHIDD  32                  // per-head dim
#define NFEAT  65536
#define NCLASS 10
#define HD     (HH * NHIDD)        // 384
#define MP     112                 // NN padded to 16 (intermediate buffers)
#define MT     7                   // MP/16
#define NT     (HD / 16)           // 24
#define ALPHA_ 0.2f
#define SLOPE_ 0.2f

// ---- stage-A GEMM blocking ---------------------------------------------
#define MB     7                   // M-tiles per workgroup (7 waves = 112 rows)
#define NTHR   (MB * 32)           // 224 threads per block
#define NBW    8                   // N-tiles per wave (128 cols)
#define WCOLS  (NBW * 16)          // 128
#define NBLK   (HD / WCOLS)        // 3 N-blocks
#define KSPLIT 128
#define KC     (NFEAT / KSPLIT)    // 512 K per slice
#define KSTEP  32                  // K-chunk staged in LDS
#define NCHUNK (KC / KSTEP)        // 16
#define PP     (KSTEP / 2)         // 16 pair-rows
#define PSTR   (2 * WCOLS + 32)    // 288 floats: disjoint bank windows

// ---- workspace layout (floats) ----------------------------------------
#define SZ_MAT   (MP * HD)                  // 43008 padded (MP x HD)
#define OFF_PART 0
#define SZ_PART  (KSPLIT * SZ_MAT)
#define OFF_XW1  (SZ_PART)
#define OFF_ES1  (OFF_XW1 + SZ_MAT)
#define OFF_ED1  (OFF_ES1 + NN * HH)
#define OFF_X1   (OFF_ED1 + NN * HH)
#define OFF_XW2  (OFF_X1 + SZ_MAT)
#define OFF_ES2  (OFF_XW2 + SZ_MAT)
#define OFF_ED2  (OFF_ES2 + NN * HH)
#define OFF_X2   (OFF_ED2 + NN * HH)
#define OFF_ATT2 (OFF_X2 + SZ_MAT)
#define OFF_ADJ  (OFF_ATT2 + HH * NN * NN)
#define OFF_H0   (OFF_ADJ + HH * MP * MP)
#define OFF_H1   (OFF_H0 + SZ_MAT)
#define OFF_ZPG  (OFF_H1 + SZ_MAT)         // zero page for padded A rows
#define OFF_END  (OFF_ZPG + NFEAT)

typedef __attribute__((ext_vector_type(2))) float v2f;
typedef __attribute__((ext_vector_type(8))) float v8f;

static __device__ __forceinline__ v8f wmma_f32(v2f a, v2f b, v8f c) {
  // (neg_a, A, neg_b, B, c_mod, C, reuse_a, reuse_b)
  return __builtin_amdgcn_wmma_f32_16x16x4_f32(false, a, false, b, (short)0, c,
                                               false, false);
}

// ---- zero workspace region ---------------------------------------------
__global__ void zero_f(float* p, int n) {
  int i = blockIdx.x * blockDim.x + threadIdx.x;
  if (i < n) p[i] = 0.0f;
}

// ---- stage A: split-K WMMA GEMM  part[ks] = x @ W1 (K-slice) -----------
// 7 waves/WG = 7 M-tiles (112 rows); each wave owns 8 N-tiles (128 cols).
// B staged through double-buffered LDS in K-pair-interleaved layout:
// element (k,c) at (k>>1)*PSTR + 2c + (k&1)  => each WMMA B operand is one
// conflict-free ds_load_b64, no repack movs.  Padded M rows read a zero page
// instead of mask-multiplying.
__global__ __launch_bounds__(NTHR) void gemm1_splitk(
    const float* __restrict__ x, const float* __restrict__ W1,
    const float* __restrict__ zpg, float* __restrict__ part) {
  __shared__ float Bs[2][PP * PSTR];      // 2 x 18 KB

  const int tid  = threadIdx.x;
  const int wave = tid >> 5;              // = M-tile within block
  const int lane = tid & 31;
  const int nb   = blockIdx.x % NBLK;     // 128-col block
  const int ks   = blockIdx.x / NBLK;     // K slice
  const int half = lane >> 4, l15 = lane & 15;
  const int row  = wave * 16 + l15;
  const float* xrow = (row < NN) ? (x + (size_t)row * NFEAT) : zpg;
  const int colbase = nb * WCOLS;
  const int kbase   = ks * KC;

  v8f acc[NBW];
  #pragma unroll
  for (int j = 0; j < NBW; ++j) acc[j] = (v8f){};

  // cooperative B loader: 512 pair-tasks (16 pair-rows x 32 col4-groups);
  // each task: 2 global b128 (rows 2k2, 2k2+1) -> 4 interleaved ds_store_b64
  auto loadB = [&](int buf, int chunk) {
    for (int pt = tid; pt < PP * (WCOLS / 4); pt += NTHR) {
      int k2 = pt >> 5;                   // pair-row 0..15
      int c4 = (pt & 31) << 2;            // col group 0,4,..,124
      const float* g =
          &W1[(size_t)(kbase + chunk * KSTEP + 2 * k2) * HD + colbase + c4];
      float4 lo = *(const float4*)g;
      float4 hi = *(const float4*)(g + HD);
      float* dp = &Bs[buf][k2 * PSTR + (c4 << 1)];
      ((float2*)dp)[0] = make_float2(lo.x, hi.x);
      ((float2*)dp)[1] = make_float2(lo.y, hi.y);
      ((float2*)dp)[2] = make_float2(lo.z, hi.z);
      ((float2*)dp)[3] = make_float2(lo.w, hi.w);
    }
  };

  loadB(0, 0);
  __syncthreads();

  for (int chunk = 0; chunk < NCHUNK; ++chunk) {
    const int buf = chunk & 1;
    if (chunk + 1 < NCHUNK) {
      loadB(buf ^ 1, chunk + 1);
      __builtin_prefetch(xrow + kbase + (chunk + 1) * KSTEP, 0, 0);
    }
    const int kk0 = kbase + chunk * KSTEP;
    #pragma unroll
    for (int kk = 0; kk < KSTEP; kk += 4) {
      const int ka = kk + 2 * half;       // even
      v2f a;
      a.x = xrow[kk0 + ka];
      a.y = xrow[kk0 + ka + 1];
      const float* bp = &Bs[buf][(ka >> 1) * PSTR + 2 * l15];
      #pragma unroll
      for (int j = 0; j < NBW; ++j) {
        v2f b = *(const v2f*)(bp + 32 * j);   // one ds_load_b64
        acc[j] = wmma_f32(a, b, acc[j]);
      }
    }
    __syncthreads();
  }

  float* pout = part + (size_t)ks * SZ_MAT + colbase + l15;
  #pragma unroll
  for (int j = 0; j < NBW; ++j) {
    #pragma unroll
    for (int r = 0; r < 8; ++r) {
      const int m = wave * 16 + r + 8 * half;
      pout[m * HD + 16 * j] = acc[j][r];
    }
  }
}

// fixed-order deterministic split-K reduction (float4 vectorized)
__global__ void reduce_part(const float* __restrict__ part,
                            float* __restrict__ xw) {
  int idx = blockIdx.x * blockDim.x + threadIdx.x;
  if (idx >= NN * (HD / 4)) return;
  const float4* p = (const float4*)part + idx;
  float4 s = make_float4(0.f, 0.f, 0.f, 0.f);
  for (int ks = 0; ks < KSPLIT; ++ks) {
    float4 v = p[(size_t)ks * (SZ_MAT / 4)];
    s.x += v.x; s.y += v.y; s.z += v.z; s.w += v.w;
  }
  ((float4*)xw)[idx] = s;
}

// ---- attention scores: es/ed per (node, head) ---------------------------
__global__ void attn_scores(const float* __restrict__ xw,
                            const float* __restrict__ a_src,
                            const float* __restrict__ a_dst,
                            float* __restrict__ es, float* __restrict__ ed) {
  int idx = blockIdx.x * blockDim.x + threadIdx.x;
  if (idx >= NN * HH) return;
  int n = idx / HH, h = idx % HH;
  float s = 0.0f, d = 0.0f;
  for (int k = 0; k < NHIDD; ++k) {
    float v = xw[n * HD + h * NHIDD + k];
    s += v * a_src[h * NHIDD + k];
    d += v * a_dst[h * NHIDD + k];
  }
  es[idx] = s;
  ed[idx] = d;
}

// ---- softmax over source + aggregate; one wave per (t,h) ---------------
__global__ __launch_bounds__(128) void attn_apply(
    const float* __restrict__ xw, const float* __restrict__ es,
    const float* __restrict__ ed, const float* __restrict__ bias,
    float* __restrict__ xout, float* __restrict__ att /* or nullptr */) {
  int gw = (blockIdx.x * 128 + threadIdx.x) >> 5;     // = t*HH + h
  int d  = threadIdx.x & 31;
  int t = gw / HH, h = gw % HH;
  float edt = ed[t * HH + h];
  float mx = -1e30f;
  for (int s = 0; s < NN; ++s) {
    float e = es[s * HH + h] + edt;
    e = e > 0.0f ? e : SLOPE_ * e;
    mx = fmaxf(mx, e);
  }
  float den = 0.0f, acc = 0.0f;
  for (int s = 0; s < NN; ++s) {
    float e = es[s * HH + h] + edt;
    e = e > 0.0f ? e : SLOPE_ * e;
    float w = __expf(e - mx);
    den += w;
    acc += w * xw[s * HD + h * NHIDD + d];
  }
  float inv = 1.0f / den;
  xout[t * HD + h * NHIDD + d] = acc * inv + bias[h * NHIDD + d];
  if (att) {                                   // store att2[h][s][t]
    for (int s = d; s < NN; s += 32) {
      float e = es[s * HH + h] + edt;
      e = e > 0.0f ? e : SLOPE_ * e;
      att[h * NN * NN + s * NN + t] = __expf(e - mx) * inv;
    }
  }
}

// ---- generic WMMA GEMM: C(MPxHD) = A(MPxHD padded) @ B(HDxHD) ----------
__global__ __launch_bounds__(128) void gemm_x_w(const float* __restrict__ A,
                                                const float* __restrict__ B,
                                                float* __restrict__ C) {
  int gw   = (blockIdx.x * 128 + threadIdx.x) >> 5;
  int lane = threadIdx.x & 31;
  int mt = gw % MT, nt = gw / MT;
  int half = lane >> 4, l15 = lane & 15;
  int row = mt * 16 + l15;
  int col = nt * 16 + l15;
  v8f acc = {};
  for (int k0 = 2 * half; k0 < HD; k0 += 4) {
    v2f a; a.x = A[row * HD + k0];     a.y = A[row * HD + k0 + 1];
    v2f b; b.x = B[k0 * HD + col];     b.y = B[(k0 + 1) * HD + col];
    acc = wmma_f32(a, b, acc);
  }
  #pragma unroll
  for (int r = 0; r < 8; ++r)
    C[(mt * 16 + r + 8 * half) * HD + col] = acc[r];
}

// ---- rebuild scrambled adjacency: adj[h][i][j] = att2[h][SRC[k]][TGT[k]]
__global__ void build_adj(const float* __restrict__ att2,
                          float* __restrict__ adj) {
  int idx = blockIdx.x * blockDim.x + threadIdx.x;
  if (idx >= HH * NN * NN) return;
  int h = idx / (NN * NN);
  int k = idx % (NN * NN);
  int s, t;
  if (k < NN * (NN - 1)) {
    s = k / (NN - 1);
    int r = k % (NN - 1);
    t = r + (r >= s ? 1 : 0);
  } else {
    s = k - NN * (NN - 1);
    t = s;
  }
  int i = k / NN, j = k % NN;
  adj[h * MP * MP + i * MP + j] = att2[h * NN * NN + s * NN + t];
}

// ---- propagation: hout = relu((1-a)*adj_h @ hin + a*x2), WMMA ----------
__global__ __launch_bounds__(128) void prop_step(const float* __restrict__ adj,
                                                 const float* __restrict__ hin,
                                                 const float* __restrict__ x2,
                                                 float* __restrict__ hout) {
  int gw   = (blockIdx.x * 128 + threadIdx.x) >> 5;
  int lane = threadIdx.x & 31;
  int mt = gw % MT, nt = gw / MT;
  int half = lane >> 4, l15 = lane & 15;
  int row = mt * 16 + l15;
  int col = nt * 16 + l15;
  v8f acc = {};
  for (int k0 = 2 * half; k0 < MP; k0 += 4) {
    v2f a; a.x = adj[row * MP + k0];   a.y = adj[row * MP + k0 + 1];
    v2f b; b.x = hin[k0 * HD + col];   b.y = hin[(k0 + 1) * HD + col];
    acc = wmma_f32(a, b, acc);
  }
  #pragma unroll
  for (int r = 0; r < 8; ++r) {
    int m = mt * 16 + r + 8 * half;
    float v = (1.0f - ALPHA_) * acc[r] + ALPHA_ * x2[m * HD + col];
    hout[m * HD + col] = fmaxf(v, 0.0f);
  }
}

// ---- output head: out = h @ Wo + bo ------------------------------------
__global__ void final_gemm(const float* __restrict__ hfin,
                           const float* __restrict__ Wo,
                           const float* __restrict__ bo,
                           float* __restrict__ out) {
  int idx = blockIdx.x * blockDim.x + threadIdx.x;
  if (idx >= NN * NCLASS) return;
  int t = idx / NCLASS, c = idx % NCLASS;
  float s = bo[c];
  for (int k = 0; k < HD; ++k) s += hfin[t * HD + k] * Wo[k * NCLASS + c];
  out[idx] = s;
}

extern "C" void kernel_launch(void* const* d_in, const int* in_sizes, int n_in,
                              void* d_out, int out_size, void* d_ws,
                              size_t ws_size, hipStream_t stream) {
  (void)in_sizes; (void)n_in; (void)out_size; (void)ws_size;
  const float* x      = (const float*)d_in[0];
  const float* W1     = (const float*)d_in[1];
  const float* a_src1 = (const float*)d_in[2];
  const float* a_dst1 = (const float*)d_in[3];
  const float* b1     = (const float*)d_in[4];
  const float* W2     = (const float*)d_in[5];
  const float* a_src2 = (const float*)d_in[6];
  const float* a_dst2 = (const float*)d_in[7];
  const float* b2     = (const float*)d_in[8];
  const float* Wo     = (const float*)d_in[9];
  const float* bo     = (const float*)d_in[10];
  float* out = (float*)d_out;
  float* ws  = (float*)d_ws;

  float* part = ws + OFF_PART;
  float* xw1  = ws + OFF_XW1;
  float* es1  = ws + OFF_ES1;
  float* ed1  = ws + OFF_ED1;
  float* x1   = ws + OFF_X1;
  float* xw2  = ws + OFF_XW2;
  float* es2  = ws + OFF_ES2;
  float* ed2  = ws + OFF_ED2;
  float* x2   = ws + OFF_X2;
  float* att2 = ws + OFF_ATT2;
  float* adj  = ws + OFF_ADJ;
  float* h0   = ws + OFF_H0;
  float* h1   = ws + OFF_H1;
  float* zpg  = ws + OFF_ZPG;

  // re-zero everything past the (fully overwritten) split-K partials
  {
    int n = OFF_END - OFF_XW1;
    zero_f<<<(n + 255) / 256, 256, 0, stream>>>(ws + OFF_XW1, n);
  }

  // stage A: LDS-tiled split-K WMMA GEMM (384 WGs x 7 waves)
  gemm1_splitk<<<NBLK * KSPLIT, NTHR, 0, stream>>>(x, W1, zpg, part);
  reduce_part<<<(NN * (HD / 4) + 255) / 256, 256, 0, stream>>>(part, xw1);

  // GAT layer 1
  attn_scores<<<(NN * HH + 255) / 256, 256, 0, stream>>>(xw1, a_src1, a_dst1,
                                                         es1, ed1);
  attn_apply<<<(NN * HH) / 4, 128, 0, stream>>>(xw1, es1, ed1, b1, x1,
                                                (float*)nullptr);

  // GAT layer 2 (keep attention)
  gemm_x_w<<<(MT * NT) / 4, 128, 0, stream>>>(x1, W2, xw2);
  attn_scores<<<(NN * HH + 255) / 256, 256, 0, stream>>>(xw2, a_src2, a_dst2,
                                                         es2, ed2);
  attn_apply<<<(NN * HH) / 4, 128, 0, stream>>>(xw2, es2, ed2, b2, x2, att2);

  // scrambled per-head adjacency
  build_adj<<<(HH * NN * NN + 255) / 256, 256, 0, stream>>>(att2, adj);

  // 12 sequential propagation steps (WMMA, fused relu epilogue)
  const float* hin = x2;
  for (int i = 0; i < HH; ++i) {
    float* hout = (i & 1) ? h1 : h0;
    prop_step<<<(MT * NT) / 4, 128, 0, stream>>>(adj + (size_t)i * MP * MP,
                                                 hin, x2, hout);
    hin = hout;
  }

  // classifier head
  final_gemm<<<(NN * NCLASS + 127) / 128, 128, 0, stream>>>(hin, Wo, bo, out);
}